// PredLayer_59519656788138
// MI455X (gfx1250) — compile-verified
//
#include <hip/hip_runtime.h>
#include <math.h>

typedef __attribute__((ext_vector_type(2))) float v2f;
typedef __attribute__((ext_vector_type(8))) float v8f;

#define D_MODEL 1024
#define VOCAB   50257
#define N_TOK   2048
#define PAD_ID  1
#define CONF    0.9f
#define SMOOTH  0.1f
#define KC      64          // K-chunk staged in LDS per buffer
#define NCHUNK  (D_MODEL / KC)

// ---------------------------------------------------------------------------
// Kernel 1: scores = x @ W^T + b  via V_WMMA_F32_16X16X4_F32
//
// grid = (ceil(V/16), N_TOK/256), block = 256 (8 waves).
// Wave w owns rows [blockIdx.y*256 + w*32, +32), cols [blockIdx.x*16, +16).
//
// W tile (16 cols x KC) is staged once per block into LDS with
// global_load_async_to_lds_b64, pre-scattered into WMMA B-fragment order:
//   element W[col_base+row][k0 + s*4 + h*2 + j]  (h = lane>=16, j = slot)
//   -> LDS byte  buf*4096 + s*256 + (h*16 + row)*8 + j*4
// so each wave's per-step B read is ds_load_b64 at s*256 + lane*8
// (conflict-free, 64 banks fully used). Double-buffered across chunks,
// register ping-pong (groups of 4 K-steps) inside a chunk.
// ---------------------------------------------------------------------------
__global__ __launch_bounds__(256)
void predlayer_gemm_wmma(const float* __restrict__ x,
                         const float* __restrict__ W,
                         const float* __restrict__ bias,
                         float* __restrict__ scores)
{
    __shared__ float bsm[2 * KC * 16];      // 2 x 4KB fragment-ordered B buffers

    const int tid  = threadIdx.x;
    const int lane = tid & 31;
    const int wave = tid >> 5;
    const int col_base = blockIdx.x * 16;
    const int m_base   = blockIdx.y * 256 + wave * 32;

    const int lm = lane & 15;               // tile row (A) / tile col (B,C)
    const int lk = (lane >> 4) * 2;         // K sub-offset for this half-wave

    // A fragment pointers: x rows for the two 16-row sub-tiles
    const float* aptr0 = x + (size_t)(m_base      + lm) * D_MODEL + lk;
    const float* aptr1 = x + (size_t)(m_base + 16 + lm) * D_MODEL + lk;

    // --- async-copy bookkeeping: 512 b64 copies per chunk, 2 per thread ---
    // copy q (0..511): row = q>>5 (0..15), p = q&31 (K-pair index)
    const int p    = tid & 31;
    const int row0 = tid >> 5;              // 0..7   (q = tid)
    const int row1 = row0 + 8;              // 8..15  (q = tid + 256)
    int wr0 = col_base + row0; if (wr0 >= VOCAB) wr0 = VOCAB - 1;
    int wr1 = col_base + row1; if (wr1 >= VOCAB) wr1 = VOCAB - 1;
    const float* gp0 = W + (size_t)wr0 * D_MODEL + p * 2;
    const float* gp1 = W + (size_t)wr1 * D_MODEL + p * 2;
    // fragment-order LDS destination (bytes) within a buffer
    const unsigned dst0 = (unsigned)((p >> 1) * 256 + ((p & 1) * 16 + row0) * 8);
    const unsigned dst1 = dst0 + 64;        // row +8 -> +64 bytes
    const unsigned smbase = (unsigned)(unsigned long long)&bsm[0];

    auto issue_chunk = [&](int chunk, int buf) {
        const int k0 = chunk * KC;
        unsigned d0 = smbase + buf * 4096 + dst0;
        unsigned d1 = smbase + buf * 4096 + dst1;
        unsigned long long g0 = (unsigned long long)(gp0 + k0);
        unsigned long long g1 = (unsigned long long)(gp1 + k0);
        asm volatile("global_load_async_to_lds_b64 %0, %1, off"
                     :: "v"(d0), "v"(g0) : "memory");
        asm volatile("global_load_async_to_lds_b64 %0, %1, off"
                     :: "v"(d1), "v"(g1) : "memory");
    };

    v8f c0 = {};
    v8f c1 = {};
    v2f a0f[2][4], a1f[2][4], bf[2][4];     // ping-pong fragment groups

    auto load_group = [&](int sbase, int k0, int buf, int which) {
        #pragma unroll
        for (int s = 0; s < 4; ++s) {
            const int st = sbase + s;
            a0f[which][s] = *(const v2f*)(aptr0 + k0 + st * 4);
            a1f[which][s] = *(const v2f*)(aptr1 + k0 + st * 4);
            bf [which][s] = *(const v2f*)(bsm + buf * 1024 + st * 64 + lane * 2);
        }
    };
    auto compute_group = [&](int which) {
        #pragma unroll
        for (int s = 0; s < 4; ++s) {
            c0 = __builtin_amdgcn_wmma_f32_16x16x4_f32(false, a0f[which][s], false,
                    bf[which][s], (short)0, c0, false, false);
            c1 = __builtin_amdgcn_wmma_f32_16x16x4_f32(false, a1f[which][s], false,
                    bf[which][s], (short)0, c1, false, false);
        }
    };

    // prologue: stage chunk 0
    issue_chunk(0, 0);
    asm volatile("s_wait_asynccnt 0x0" ::: "memory");
    __syncthreads();

    for (int c = 0; c < NCHUNK; ++c) {
        const int buf = c & 1;
        const int k0  = c * KC;
        if (c + 1 < NCHUNK) issue_chunk(c + 1, buf ^ 1);   // overlap with compute

        load_group(0,  k0, buf, 0);
        load_group(4,  k0, buf, 1);
        compute_group(0);
        load_group(8,  k0, buf, 0);
        compute_group(1);
        load_group(12, k0, buf, 1);
        compute_group(0);
        compute_group(1);

        asm volatile("s_wait_asynccnt 0x0" ::: "memory");  // own copies done
        __syncthreads();                                    // all waves' copies done
    }

    const int col = col_base + lm;
    if (col >= VOCAB) return;               // ragged tail: drop these lanes' stores
    const float bv = bias[col];

    // C/D layout: lane l, vgpr r -> M = (l>>4)*8 + r, N = l%16
    const int mrow = m_base + (lane >> 4) * 8;
    #pragma unroll
    for (int r = 0; r < 8; ++r) {
        scores[(size_t)(mrow + r)      * VOCAB + col] = c0[r] + bv;
        scores[(size_t)(mrow + 16 + r) * VOCAB + col] = c1[r] + bv;
    }
}

// ---------------------------------------------------------------------------
// Kernel 2: per-row smoothed-KL loss. One block per row (2048 blocks).
//   loss_row = [conf*log conf + 0.1*log sv]
//            - [ sv*(sum - s_label - s_pad) + conf*s_label - lse ]
// ---------------------------------------------------------------------------
__global__ __launch_bounds__(256)
void predlayer_row_loss(const float* __restrict__ scores,
                        const int* __restrict__ labels,
                        float* __restrict__ row_loss)
{
    __shared__ float sh_max[256];
    __shared__ float sh_sum[256];

    const int row = blockIdx.x;
    const int tid = threadIdx.x;
    const float* s = scores + (size_t)row * VOCAB;

    float lmax = -INFINITY;
    float lsum = 0.0f;
    for (int i = tid; i < VOCAB; i += 256) {
        float v = s[i];
        lmax = fmaxf(lmax, v);
        lsum += v;
    }
    sh_max[tid] = lmax;
    sh_sum[tid] = lsum;
    __syncthreads();
    for (int off = 128; off > 0; off >>= 1) {
        if (tid < off) {
            sh_max[tid] = fmaxf(sh_max[tid], sh_max[tid + off]);
            sh_sum[tid] += sh_sum[tid + off];
        }
        __syncthreads();
    }
    const float rmax = sh_max[0];
    const float rsum = sh_sum[0];
    __syncthreads();

    float lexp = 0.0f;
    for (int i = tid; i < VOCAB; i += 256) {
        lexp += expf(s[i] - rmax);
    }
    sh_max[tid] = lexp;
    __syncthreads();
    for (int off = 128; off > 0; off >>= 1) {
        if (tid < off) sh_max[tid] += sh_max[tid + off];
        __syncthreads();
    }

    if (tid == 0) {
        const float lse = rmax + logf(sh_max[0]);
        const int label = labels[row];
        float loss = 0.0f;
        if (label != PAD_ID) {
            const float sv    = SMOOTH / (float)(VOCAB - 2);
            const float slab  = s[label];
            const float spad  = s[PAD_ID];
            const float plogp = CONF * logf(CONF) + SMOOTH * logf(sv);
            const float plogq = sv * (rsum - slab - spad) + CONF * slab - lse;
            loss = plogp - plogq;
        }
        row_loss[row] = loss;
    }
}

// ---------------------------------------------------------------------------
// Kernel 3: reduce 2048 row losses -> loss / N  (single block, deterministic)
// ---------------------------------------------------------------------------
__global__ __launch_bounds__(256)
void predlayer_final(const float* __restrict__ row_loss,
                     float* __restrict__ out_loss)
{
    __shared__ float sh[256];
    const int tid = threadIdx.x;
    float acc = 0.0f;
    for (int i = tid; i < N_TOK; i += 256) acc += row_loss[i];
    sh[tid] = acc;
    __syncthreads();
    for (int off = 128; off > 0; off >>= 1) {
        if (tid < off) sh[tid] += sh[tid + off];
        __syncthreads();
    }
    if (tid == 0) out_loss[0] = sh[0] / (float)N_TOK;
}

// ---------------------------------------------------------------------------
extern "C" void kernel_launch(void* const* d_in, const int* in_sizes, int n_in,
                              void* d_out, int out_size, void* d_ws, size_t ws_size,
                              hipStream_t stream)
{
    const float* x      = (const float*)d_in[0];
    const float* W      = (const float*)d_in[1];
    const float* b      = (const float*)d_in[2];
    const int*   labels = (const int*)d_in[3];

    float* scores   = (float*)d_out;                       // [N_TOK, VOCAB]
    float* loss_out = scores + (size_t)N_TOK * VOCAB;      // scalar
    float* row_loss = (float*)d_ws;                        // [N_TOK] scratch

    dim3 gemm_grid((VOCAB + 15) / 16, N_TOK / 256);
    predlayer_gemm_wmma<<<gemm_grid, 256, 0, stream>>>(x, W, b, scores);

    predlayer_row_loss<<<N_TOK, 256, 0, stream>>>(scores, labels, row_loss);

    predlayer_final<<<1, 256, 0, stream>>>(row_loss, loss_out);
}